// LDE1D_38439957299392
// MI455X (gfx1250) — compile-verified
//
#include <hip/hip_runtime.h>

// Problem constants (from reference setup_inputs)
#define B_   64
#define T_   4096
#define D_   256
#define K_   64
#define CH_  4        // T-chunks (deterministic chunk partials, no atomics)
#define TT_  32       // t-rows per tile staged in LDS
#define XPAD 260      // LDS row stride (floats): 256 data + 4 pad, TDM pad feature matches
#define MPAD 260
#define WPAD 80       // LDS row stride for w (2-row offset = 160 % 64 = 32 -> disjoint half-waves)
#define NTILE ((T_ / CH_) / TT_)   // 32 tiles per block

typedef __attribute__((ext_vector_type(2))) float v2f;
typedef __attribute__((ext_vector_type(8))) float v8f;
typedef __attribute__((ext_vector_type(4))) unsigned int v4u;
typedef __attribute__((ext_vector_type(8))) int v8i;
typedef __attribute__((ext_vector_type(4))) int v4i;

// LDS byte offset of a __shared__ object (ptrtoint of addrspace(3) pointer)
__device__ __forceinline__ unsigned lds_addr_of(const void* p) {
    return (unsigned)(unsigned long long)(const __attribute__((address_space(3))) char*)p;
}

// TDM: DMA a [rows x 256] f32 tile global->LDS, HW-padding each 256-DWORD row
// by 4 DWORDs (pad_interval code 7 = 256 DWORDs, pad_amount code 3 = 4 DWORDs)
// so the LDS image lands with row stride 260 floats. D# per ISA ch.8.
__device__ __forceinline__ void tdm_load_2d_padded(unsigned lds_off,
                                                   const void* gptr, int rows) {
    const unsigned long long ga = (unsigned long long)gptr;
    v4u g0;
    g0[0] = 1u;                                           // count=1, no gather
    g0[1] = lds_off;                                      // lds_addr
    g0[2] = (unsigned)(ga & 0xFFFFFFFFu);                 // global_addr[31:0]
    g0[3] = (unsigned)((ga >> 32) & 0x01FFFFFFu)          // global_addr[56:32]
          | (2u << 30);                                   // type=2 (image)
    v8i g1;
    g1[0] = (2 << 16)          // data_size = 4B
          | (1 << 20)          // pad_enable
          | (7 << 22)          // pad_interval: 256 DWORDs
          | (3 << 25);         // pad_amount:   4 DWORDs
    g1[1] = (int)(256u << 16);                 // tensor_dim0 (lo16 at [63:48])
    g1[2] = (int)(((unsigned)rows) << 16);     // tensor_dim1 (lo16 at [95:80])
    g1[3] = (int)(256u << 16);                 // tile_dim0   at [127:112]
    g1[4] = rows;                              // tile_dim1; tile_dim2=0
    g1[5] = 256;                               // tensor_dim0_stride (lo32)
    g1[6] = 0;                                 // dim0_stride hi, dim1_stride lo
    g1[7] = 0;
    const v4i z4 = {0, 0, 0, 0};               // groups 2/3 unused (2-D tile)
    const v8i z8 = {0, 0, 0, 0, 0, 0, 0, 0};   // extra group (clang-23 6-arg form)
    __builtin_amdgcn_tensor_load_to_lds(g0, g1, z4, z4, z8, 0);
}

// ---------------------------------------------------------------------------
// Pass 1: per (b, chunk) block computes chunk-partial Sxw[k,d] and Sw[k]
//   GEMM1 (logits):  xm[t,k] = sum_d x[t,d]*mu[k,d]      via v_wmma_f32_16x16x4_f32
//   softmax over K (=64), scaled by weights[b,t]
//   GEMM2 (moments): Sxw[k,d] += sum_t w[t,k]*x[t,d]     via v_wmma_f32_16x16x4_f32
// x tiles double-buffered; TDM prefetches tile i+1 under tile i's compute.
// ---------------------------------------------------------------------------
__global__ __launch_bounds__(256) void lde_pass1(
    const float* __restrict__ x, const float* __restrict__ wts,
    const float* __restrict__ mu, const float* __restrict__ s,
    float* __restrict__ sxwp, float* __restrict__ swp)
{
    __shared__ __align__(16) float xs[2][TT_][XPAD];  // x tiles (double buffer)
    __shared__ __align__(16) float ms[K_][MPAD];      // mu (whole)
    __shared__ float wls[TT_][WPAD];                  // logits -> softmax weights
    __shared__ float x2s[TT_];
    __shared__ float mu2s[K_];
    __shared__ float sls[K_];

    const int tid  = threadIdx.x;
    const int lane = tid & 31;
    const int wv   = tid >> 5;              // 0..7 waves
    const int b    = blockIdx.x / CH_;
    const int ch   = blockIdx.x % CH_;
    const int tcBeg = ch * (T_ / CH_);      // 1024 t-rows per chunk

    // ---- TDM: stage mu and the first x tile into LDS (HW row padding) ----
    if (tid == 0) {
        tdm_load_2d_padded(lds_addr_of(&ms[0][0]), mu, K_);
        tdm_load_2d_padded(lds_addr_of(&xs[0][0][0]),
                           x + ((size_t)b * T_ + tcBeg) * D_, TT_);
    }
    if (tid < K_) sls[tid] = s[tid];
    __builtin_amdgcn_s_wait_tensorcnt(0);   // no-op for waves with TENSORcnt==0
    __syncthreads();
    if (tid < K_) {                         // ||mu_k||^2
        float acc = 0.f;
        for (int d = 0; d < D_; ++d) { const float v = ms[tid][d]; acc += v * v; }
        mu2s[tid] = acc;
    }

    // GEMM2 register accumulators: wave -> k-tile kt2 (0..3), d-tile group dg (0..1) of 8
    const int kt2 = wv & 3;
    const int dg  = wv >> 2;
    v8f acc2[8];
    #pragma unroll
    for (int j = 0; j < 8; ++j) acc2[j] = (v8f)0.0f;
    float swacc = 0.f;

    // GEMM1 tile assignment: 8 waves cover [2 t-tiles x 4 k-tiles]
    const int trow = wv & 1;
    const int kcol = wv >> 1;

    for (int tile = 0; tile < NTILE; ++tile) {
        const int t0  = tcBeg + tile * TT_;
        const int buf = tile & 1;

        // ---- TDM prefetch of next tile into the other buffer ----
        if (tid == 0 && (tile + 1) < NTILE) {
            tdm_load_2d_padded(lds_addr_of(&xs[buf ^ 1][0][0]),
                               x + ((size_t)b * T_ + t0 + TT_) * D_, TT_);
        }
        if (tid < TT_) {                    // ||x_t||^2
            float acc = 0.f;
            for (int d = 0; d < D_; ++d) { const float v = xs[buf][tid][d]; acc += v * v; }
            x2s[tid] = acc;
        }
        __syncthreads();

        // ---- GEMM1: 16x16 xm tile per wave, K=4 steps over D ----
        v8f c1 = (v8f)0.0f;
        {
            const int tA    = trow * 16 + (lane & 15);      // A: M index
            const int kB    = kcol * 16 + (lane & 15);      // B: N index
            const int khalf = (lane >> 4) * 2;              // K split across half-waves
            for (int dstep = 0; dstep < D_ / 4; ++dstep) {  // 64 WMMAs
                const int d0 = dstep * 4 + khalf;
                const v2f a  = *(const v2f*)&xs[buf][tA][d0];   // ds_load_b64
                const v2f bb = *(const v2f*)&ms[kB][d0];        // ds_load_b64
                c1 = __builtin_amdgcn_wmma_f32_16x16x4_f32(
                        false, a, false, bb, (short)0, c1, false, false);
            }
        }
        // ---- logits -> LDS  (C layout: N=lane&15, M=v+8*(lane>>4)) ----
        {
            const int kk   = kcol * 16 + (lane & 15);
            const float m2 = mu2s[kk];
            const float sk = sls[kk];
            const int thi  = trow * 16 + 8 * (lane >> 4);
            #pragma unroll
            for (int v = 0; v < 8; ++v) {
                const int tt = thi + v;
                const float dist = x2s[tt] - 2.0f * c1[v] + m2;
                wls[tt][kk] = -sk * dist;
            }
        }
        __syncthreads();

        // ---- softmax over K=64, scaled by weights[b,t] ----
        if (tid < TT_) {
            const int r = tid;
            float mx = -3.4e38f;
            for (int k = 0; k < K_; ++k) mx = fmaxf(mx, wls[r][k]);
            float sum = 0.f;
            for (int k = 0; k < K_; ++k) {
                const float e = __expf(wls[r][k] - mx);
                wls[r][k] = e; sum += e;
            }
            const float scale = wts[(size_t)b * T_ + t0 + r] / sum;
            for (int k = 0; k < K_; ++k) wls[r][k] *= scale;
        }
        __syncthreads();

        // ---- Sw partial: thread k sums its column ----
        if (tid < K_) {
            float a = 0.f;
            for (int r = 0; r < TT_; ++r) a += wls[r][tid];
            swacc += a;
        }

        // ---- GEMM2: Sxw[k,d] += w^T x, K=4 steps over the 32 t-rows ----
        {
            const int kA    = kt2 * 16 + (lane & 15);       // A: M index (=k)
            const int tHalf = (lane >> 4) * 2;
            for (int ts = 0; ts < TT_ / 4; ++ts) {          // 8 x 8 = 64 WMMAs
                const int tt0 = ts * 4 + tHalf;
                v2f a;
                a.x = wls[tt0][kA]; a.y = wls[tt0 + 1][kA];
                #pragma unroll
                for (int j = 0; j < 8; ++j) {
                    const int dbase = (dg * 8 + j) * 16 + (lane & 15);
                    v2f bb;
                    bb.x = xs[buf][tt0][dbase]; bb.y = xs[buf][tt0 + 1][dbase];
                    acc2[j] = __builtin_amdgcn_wmma_f32_16x16x4_f32(
                                false, a, false, bb, (short)0, acc2[j], false, false);
                }
            }
        }

        // ---- drain the prefetch before switching buffers ----
        __builtin_amdgcn_s_wait_tensorcnt(0);
        __syncthreads();
    }

    // ---- write chunk partials (unique writer per element -> deterministic) ----
    {
        float* dst = sxwp + (size_t)ch * (B_ * K_ * D_) + (size_t)b * (K_ * D_);
        const int dlo = lane & 15;
        const int khi = 8 * (lane >> 4);
        #pragma unroll
        for (int j = 0; j < 8; ++j) {
            const int d = (dg * 8 + j) * 16 + dlo;
            #pragma unroll
            for (int v = 0; v < 8; ++v) {
                const int k = kt2 * 16 + khi + v;
                dst[(size_t)k * D_ + d] = acc2[j][v];
            }
        }
    }
    if (tid < K_) swp[(size_t)ch * (B_ * K_) + b * K_ + tid] = swacc;
}

// ---------------------------------------------------------------------------
// Pass 2: reduce chunk partials in fixed order, out = Sxw/Sw - mu
// ---------------------------------------------------------------------------
__global__ __launch_bounds__(256) void lde_finalize(
    const float* __restrict__ sxwp, const float* __restrict__ swp,
    const float* __restrict__ mu, float* __restrict__ out)
{
    const size_t idx = (size_t)blockIdx.x * 256 + threadIdx.x;  // over B*K*D
    const int d  = (int)(idx & (D_ - 1));
    const int k  = (int)((idx >> 8) & (K_ - 1));
    const int bk = (int)(idx >> 8);                              // b*K + k
    float sx = 0.f, sw = 0.f;
    #pragma unroll
    for (int ch = 0; ch < CH_; ++ch) {
        sx += sxwp[(size_t)ch * (B_ * K_ * D_) + idx];
        sw += swp[ch * (B_ * K_) + bk];
    }
    out[idx] = sx / sw - mu[(size_t)k * D_ + d];
}

// ---------------------------------------------------------------------------
extern "C" void kernel_launch(void* const* d_in, const int* in_sizes, int n_in,
                              void* d_out, int out_size, void* d_ws, size_t ws_size,
                              hipStream_t stream) {
    (void)in_sizes; (void)n_in; (void)out_size; (void)ws_size;
    const float* x   = (const float*)d_in[0];   // [B,T,D] f32
    const float* wts = (const float*)d_in[1];   // [B,T]   f32
    const float* mu  = (const float*)d_in[2];   // [K,D]   f32
    const float* s   = (const float*)d_in[3];   // [K]     f32
    float* out  = (float*)d_out;                // [B,K*D] f32
    float* sxwp = (float*)d_ws;                              // CH*B*K*D floats (16 MB)
    float* swp  = sxwp + (size_t)CH_ * B_ * K_ * D_;         // CH*B*K floats

    lde_pass1<<<B_ * CH_, 256, 0, stream>>>(x, wts, mu, s, sxwp, swp);
    lde_finalize<<<(B_ * K_ * D_) / 256, 256, 0, stream>>>(sxwp, swp, mu, out);
}